// Verifier_81114752352321
// MI455X (gfx1250) — compile-verified
//
#include <hip/hip_runtime.h>
#include <math.h>

// ---------------------------------------------------------------------------
// DeepPoly-style verifier for a 784->2048->2048->2048->10 MLP (SPU activation)
// Optimized for MI455X (gfx1250):
//  - all dense back-substitution GEMMs run on V_WMMA_F32_16X16X4_F32
//  - HBM -> LDS staging uses GLOBAL_LOAD_ASYNC_TO_LDS_B128 (ASYNCcnt) with
//    double-buffered LDS so DMA overlaps WMMA compute
//  - diagonal SPU constraints handled as elementwise scale passes
// ---------------------------------------------------------------------------

#define H     2048
#define DIN   784
#define DOUT  10
#define LD1   832    // padded width of [W1|b1] constraints (785 -> 832)
#define LD2   2112   // padded width of [W|b] for 2048-wide layers (2049 -> 2112)
#define BC1   784    // bias column index in LD1-wide matrices
#define BC2   2048   // bias column index in LD2-wide matrices
#define MEAN_C      0.1307f
#define SIGMA_C     0.3081f
#define SIG_SCALE_C 5.0f

typedef float v2f __attribute__((ext_vector_type(2)));
typedef float v8f __attribute__((ext_vector_type(8)));

// ---------------------------------------------------------------- helpers ---
__device__ __forceinline__ float sigm(float x) { return 1.0f / (1.0f + expf(-x)); }
__device__ __forceinline__ float spu_f(float x) {
    return (x >= 0.0f) ? (x * x - 0.5f) : (sigm(-x) - 1.0f);
}
__device__ __forceinline__ void sig_tan(float x, float& s, float& i) {
    float sg = sigm(x);
    s = -sg * (1.0f - sg);
    i = -sg - s * x;
}

// ------------------------------------------------------------- WMMA GEMM ---
// C[M x ldp] = A[:, 0:J] @ P[J x ldp]  (all f32, fully tiled, ldc == ldp)
// 64x64 block tile, 4 waves (wave32), each wave computes a 16x64 strip.
// Double-buffered LDS staged via async global->LDS DMA (ASYNCcnt).
#define GEMM_TM 64
#define GEMM_TN 64
#define GEMM_TK 32
#define LDS_A_STRIDE 36   // 32 + 4 pad  (float4-aligned: 36*4=144B)
#define LDS_B_STRIDE 72   // 64 + 8 pad  (float4-aligned: 72*4=288B)

__global__ __launch_bounds__(128)
void wmma_gemm_kernel(const float* __restrict__ A, int lda,
                      const float* __restrict__ P, int ldp,
                      float* __restrict__ C, int ldc, int J)
{
    __shared__ float As[2][GEMM_TM * LDS_A_STRIDE];
    __shared__ float Bs[2][GEMM_TK * LDS_B_STRIDE];

    const int tid  = threadIdx.x;
    const int lane = tid & 31;
    const int wave = tid >> 5;               // 0..3
    const int bm   = blockIdx.y * GEMM_TM;
    const int bn   = blockIdx.x * GEMM_TN;

    // Per-thread staging coordinates: 4 x 16B chunks of A, 4 x 16B chunks of B.
    const float* aga[4];                 // global source (kk added per stage)
    const float* bga[4];
    unsigned     aofs[2][4], bofs[2][4]; // LDS byte addresses (both buffers)
    #pragma unroll
    for (int i = 0; i < 4; ++i) {
        int idx = tid + i * 128;
        int r   = idx >> 3, c4 = idx & 7;         // A: 8 float4 per 32-wide row
        aga[i] = A + (size_t)(bm + r) * lda + c4 * 4;
        aofs[0][i] = (unsigned)(size_t)&As[0][r * LDS_A_STRIDE + c4 * 4];
        aofs[1][i] = (unsigned)(size_t)&As[1][r * LDS_A_STRIDE + c4 * 4];
        int rb  = idx >> 4, cb4 = idx & 15;       // B: 16 float4 per 64-wide row
        bga[i] = P + (size_t)rb * ldp + bn + cb4 * 4;
        bofs[0][i] = (unsigned)(size_t)&Bs[0][rb * LDS_B_STRIDE + cb4 * 4];
        bofs[1][i] = (unsigned)(size_t)&Bs[1][rb * LDS_B_STRIDE + cb4 * 4];
    }

    // Issue one stage of async HBM->LDS DMA: 8 instructions per wave
    // (32 lanes x 16B = 512B each), tracked by ASYNCcnt.
    auto issue = [&](int kk, int buf) {
        #pragma unroll
        for (int i = 0; i < 4; ++i) {
            const float* ga = aga[i] + kk;
            asm volatile("global_load_async_to_lds_b128 %0, %1, off"
                         :: "v"(aofs[buf][i]), "v"(ga) : "memory");
            const float* gb = bga[i] + (size_t)kk * ldp;
            asm volatile("global_load_async_to_lds_b128 %0, %1, off"
                         :: "v"(bofs[buf][i]), "v"(gb) : "memory");
        }
    };

    v8f acc[4] = {};
    const int mrow = wave * 16 + (lane & 15);   // row within block tile
    const int ksel = (lane >> 4) * 2;           // 0 for lanes 0-15, 2 for 16-31

    issue(0, 0);
    const int nstages = J / GEMM_TK;
    for (int s = 0; s < nstages; ++s) {
        const int buf = s & 1;
        if (s + 1 < nstages) {
            issue((s + 1) * GEMM_TK, buf ^ 1);
            // 8 next-stage DMAs may remain in flight; current stage complete
            // (async loads complete in order within a wave).
            asm volatile("s_wait_asynccnt 0x8" ::: "memory");
        } else {
            asm volatile("s_wait_asynccnt 0x0" ::: "memory");
        }
        __syncthreads();   // all waves' DMA waits observed -> LDS tile valid

        const float* as = As[buf];
        const float* bs = Bs[buf];
        #pragma unroll
        for (int k0 = 0; k0 < GEMM_TK; k0 += 4) {
            // A fragment 16x4: lanes 0-15 hold K={k0,k0+1}, lanes 16-31 K={k0+2,k0+3}
            v2f a;
            a.x = as[mrow * LDS_A_STRIDE + k0 + ksel];
            a.y = as[mrow * LDS_A_STRIDE + k0 + ksel + 1];
            #pragma unroll
            for (int nt = 0; nt < 4; ++nt) {
                int col = nt * 16 + (lane & 15);
                v2f b;
                b.x = bs[(k0 + ksel) * LDS_B_STRIDE + col];
                b.y = bs[(k0 + ksel + 1) * LDS_B_STRIDE + col];
                acc[nt] = __builtin_amdgcn_wmma_f32_16x16x4_f32(
                    false, a, false, b, (short)0, acc[nt], false, false);
            }
        }
        __syncthreads();   // tile fully consumed before buf is re-targeted
    }

    // Store: D VGPR v -> row v + 8*(lane>>4), col = lane&15 within each 16x16 tile
    const int colt   = lane & 15;
    const int rowsel = (lane >> 4) * 8;
    #pragma unroll
    for (int nt = 0; nt < 4; ++nt) {
        int gc = bn + nt * 16 + colt;
        #pragma unroll
        for (int v = 0; v < 8; ++v) {
            int gr = bm + wave * 16 + rowsel + v;
            C[(size_t)gr * ldc + gc] = acc[nt][v];
        }
    }
}

// -------------------------------------------------------------- elementwise -
__global__ void k_input_bounds(const float* __restrict__ x, const float* __restrict__ eps,
                               float* __restrict__ ub0, float* __restrict__ lb0)
{
    int j = blockIdx.x * blockDim.x + threadIdx.x;
    if (j >= DIN) return;
    float e = eps[0];
    float u = fminf(x[j] + e, 1.0f);
    float l = fmaxf(x[j] - e, 0.0f);
    ub0[j] = (u - MEAN_C) / SIGMA_C;
    lb0[j] = (l - MEAN_C) / SIGMA_C;
}

// Build padded constraint matrix c = [W | b | 0...]
__global__ void k_build_c(const float* __restrict__ W, const float* __restrict__ b,
                          float* __restrict__ c, int rows, int D, int ld)
{
    size_t idx = (size_t)blockIdx.x * blockDim.x + threadIdx.x;
    size_t total = (size_t)rows * ld;
    if (idx >= total) return;
    int row = (int)(idx / ld);
    int col = (int)(idx % ld);
    float v = 0.0f;
    if (col < D)       v = W[(size_t)row * D + col];
    else if (col == D) v = b[row];
    c[idx] = v;
}

// Interval bounds through one affine layer (block per row)
__global__ void k_interval_affine(const float* __restrict__ W, const float* __restrict__ b,
                                  const float* __restrict__ ubp, const float* __restrict__ lbp,
                                  int D, float* __restrict__ uo, float* __restrict__ lo)
{
    __shared__ float r1[256], r2[256];
    int row = blockIdx.x, tid = threadIdx.x;
    const float* w = W + (size_t)row * D;
    float au = 0.0f, al = 0.0f;
    for (int j = tid; j < D; j += 256) {
        float wv = w[j], u = ubp[j], l = lbp[j];
        au += (wv > 0.0f) ? wv * u : wv * l;
        al += (wv > 0.0f) ? wv * l : wv * u;
    }
    r1[tid] = au; r2[tid] = al; __syncthreads();
    for (int s = 128; s > 0; s >>= 1) {
        if (tid < s) { r1[tid] += r1[tid + s]; r2[tid] += r2[tid + s]; }
        __syncthreads();
    }
    if (tid == 0) { uo[row] = r1[0] + b[row]; lo[row] = r2[0] + b[row]; }
}

// analyze_spu, elementwise
__global__ void k_spu_analyze(const float* __restrict__ ub, const float* __restrict__ lb,
                              float* __restrict__ su, float* __restrict__ iu,
                              float* __restrict__ sl, float* __restrict__ il,
                              float* __restrict__ nub, float* __restrict__ nlb, int n)
{
    int t = blockIdx.x * blockDim.x + threadIdx.x;
    if (t >= n) return;
    float ux = ub[t], lx = lb[t];
    float uy = spu_f(ux), ly = spu_f(lx);
    float sj = (uy - ly) / (ux - lx);
    float ij = uy - sj * ux;
    bool right = lx > 0.0f;
    bool left  = ux <= 0.0f;
    bool crossing = (!left) && (!right);
    float mid = 0.5f * (ux + lx);
    float sp = 2.0f * mid, ip = -mid * mid - 0.5f;
    float ssm, ism; sig_tan(mid, ssm, ism);
    float spu_s = 2.0f * ux, ipu = -ux * ux - 0.5f;
    float limit = spu_s * lx + ipu;
    float clm = (fabsf(lx) > ux) ? 1.0f : 0.0f;
    float clp = sigm((clm - 0.5f) * 2.0f * SIG_SCALE_C) * (ly - limit) + limit;
    bool ptm = clp < -0.5f;
    float Dd = 4.0f * lx * lx - 4.0f * clp - 2.0f;
    float x2 = (2.0f * lx + sqrtf(fmaxf(Dd, 0.0f))) * 0.5f;
    float spt = 2.0f * x2, ipt = -x2 * x2 - 0.5f;
    float sjn = (-0.5f - clp) / (0.0f - lx);
    float ijn = -0.5f;
    float s_cl = ptm ? spt : sjn;
    float i_cl = ptm ? ipt : ijn;
    float ssl, isl; sig_tan(lx, ssl, isl);
    float stv = ssl * ux + isl - uy;
    float L = lx, R = 0.0f;
    #pragma unroll
    for (int it = 0; it < 10; ++it) {
        float m = 0.5f * (L + R);
        float sm_, im_; sig_tan(m, sm_, im_);
        bool mask = (sm_ * ux + im_ - uy) > 0.0f;
        L = mask ? m : L;
        R = mask ? R : m;
    }
    float scp = sigm(-SIG_SCALE_C) * (L - lx) + lx;
    float ssc, isc; sig_tan(scp, ssc, isc);
    bool up_mask = stv > 0.0f;
    float s_cu = up_mask ? ssc : sj;
    float i_cu = up_mask ? isc : ij;
    float rf = right ? 1.0f : 0.0f, lf = left ? 1.0f : 0.0f, cf = crossing ? 1.0f : 0.0f;
    su[t] = rf * sj + lf * ssm + cf * s_cu;
    iu[t] = rf * ij + lf * ism + cf * i_cu;
    sl[t] = rf * sp + lf * sj + cf * s_cl;
    il[t] = rf * ip + lf * ij + cf * i_cl;
    nub[t] = fmaxf(uy, ly);
    nlb[t] = crossing ? -0.5f : fminf(uy, ly);
}

// Fused back-substitution of spu1 through [c1, input] (block per row)
__global__ void k_bs_spu1(const float* __restrict__ W1, const float* __restrict__ b1,
                          const float* __restrict__ su, const float* __restrict__ iu,
                          const float* __restrict__ sl, const float* __restrict__ il,
                          const float* __restrict__ ub0, const float* __restrict__ lb0,
                          float* __restrict__ ub, float* __restrict__ lb)
{
    __shared__ float r1[256], r2[256];
    int row = blockIdx.x, tid = threadIdx.x;
    const float* w = W1 + (size_t)row * DIN;
    float s_u = su[row], s_l = sl[row];
    float au = 0.0f, al = 0.0f;
    for (int j = tid; j < DIN; j += 256) {
        float wu = s_u * w[j];
        au += (wu > 0.0f) ? wu * ub0[j] : wu * lb0[j];
        float wl = s_l * w[j];
        al += (wl > 0.0f) ? wl * lb0[j] : wl * ub0[j];
    }
    r1[tid] = au; r2[tid] = al; __syncthreads();
    for (int s = 128; s > 0; s >>= 1) {
        if (tid < s) { r1[tid] += r1[tid + s]; r2[tid] += r2[tid + s]; }
        __syncthreads();
    }
    if (tid == 0) {
        float vu = r1[0] + s_u * b1[row] + iu[row];
        float vl = r2[0] + s_l * b1[row] + il[row];
        ub[row] = fminf(ub[row], vu);
        lb[row] = fmaxf(lb[row], vl);
    }
}

// Start of affine-layer back-substitution: push [W|b] through a SPU diag layer.
// Au[i,k] = max(W,0)*su[k]+min(W,0)*sl[k]; bias col = b + sum(max*iu+min*il).
__global__ void k_diag_from_affine(const float* __restrict__ W, const float* __restrict__ b,
                                   const float* __restrict__ su, const float* __restrict__ iu,
                                   const float* __restrict__ sl, const float* __restrict__ il,
                                   float* __restrict__ Au, float* __restrict__ Al,
                                   int ld, int rows_real)
{
    __shared__ float r1[256], r2[256];
    int row = blockIdx.x, tid = threadIdx.x;
    if (row >= rows_real) {  // zero-fill padded rows (final 10->64 layer)
        for (int c = tid; c < ld; c += 256) {
            Au[(size_t)row * ld + c] = 0.0f;
            Al[(size_t)row * ld + c] = 0.0f;
        }
        return;
    }
    const float* w = W + (size_t)row * H;
    float bu = 0.0f, bl = 0.0f;
    for (int k = tid; k < H; k += 256) {
        float wv = w[k];
        float wp = fmaxf(wv, 0.0f), wn = fminf(wv, 0.0f);
        Au[(size_t)row * ld + k] = wp * su[k] + wn * sl[k];
        Al[(size_t)row * ld + k] = wp * sl[k] + wn * su[k];
        bu += wp * iu[k] + wn * il[k];
        bl += wp * il[k] + wn * iu[k];
    }
    for (int c = H + 1 + tid; c < ld; c += 256) {   // zero pad cols
        Au[(size_t)row * ld + c] = 0.0f;
        Al[(size_t)row * ld + c] = 0.0f;
    }
    r1[tid] = bu; r2[tid] = bl; __syncthreads();
    for (int s = 128; s > 0; s >>= 1) {
        if (tid < s) { r1[tid] += r1[tid + s]; r2[tid] += r2[tid + s]; }
        __syncthreads();
    }
    if (tid == 0) {
        Au[(size_t)row * ld + H] = b[row] + r1[0];
        Al[(size_t)row * ld + H] = b[row] + r2[0];
    }
}

// Start of SPU-layer back-substitution: A = diag(s) @ cp, bias col += intercept
__global__ void k_rowscale(const float* __restrict__ cp,
                           const float* __restrict__ su, const float* __restrict__ iu,
                           const float* __restrict__ sl, const float* __restrict__ il,
                           float* __restrict__ Au, float* __restrict__ Al,
                           int ld, int biascol, size_t total)
{
    size_t idx = (size_t)blockIdx.x * blockDim.x + threadIdx.x;
    if (idx >= total) return;
    int i = (int)(idx / ld);
    int c = (int)(idx % ld);
    float v = cp[idx];
    float au = su[i] * v;
    float al = sl[i] * v;
    if (c == biascol) { au += iu[i]; al += il[i]; }
    Au[idx] = au;
    Al[idx] = al;
}

// Pass dense (A_u, A_l) through a SPU diag layer (block per row).
__global__ void k_colpass(const float* __restrict__ Au, const float* __restrict__ Al,
                          const float* __restrict__ su, const float* __restrict__ iu,
                          const float* __restrict__ sl, const float* __restrict__ il,
                          float* __restrict__ Bu, float* __restrict__ Bl, int ld)
{
    __shared__ float r1[256], r2[256];
    int row = blockIdx.x, tid = threadIdx.x;
    const float* au = Au + (size_t)row * ld;
    const float* al = Al + (size_t)row * ld;
    float* bu = Bu + (size_t)row * ld;
    float* bl = Bl + (size_t)row * ld;
    float cu = 0.0f, cl = 0.0f;
    for (int k = tid; k < H; k += 256) {
        float u = au[k];
        float up = fmaxf(u, 0.0f), un = fminf(u, 0.0f);
        bu[k] = up * su[k] + un * sl[k];
        cu += up * iu[k] + un * il[k];
        float l = al[k];
        float lp = fmaxf(l, 0.0f), ln = fminf(l, 0.0f);
        bl[k] = lp * sl[k] + ln * su[k];
        cl += lp * il[k] + ln * iu[k];
    }
    r1[tid] = cu; r2[tid] = cl; __syncthreads();
    for (int s = 128; s > 0; s >>= 1) {
        if (tid < s) { r1[tid] += r1[tid + s]; r2[tid] += r2[tid + s]; }
        __syncthreads();
    }
    if (tid == 0) {
        bu[H] = au[H] + r1[0];
        bl[H] = al[H] + r2[0];
    }
}

// After a GEMM through an affine layer: carry A's bias into C's bias column
__global__ void k_add_bias(float* __restrict__ C, int ldc, int bc,
                           const float* __restrict__ A, int lda, int ac, int M)
{
    int i = blockIdx.x * blockDim.x + threadIdx.x;
    if (i >= M) return;
    C[(size_t)i * ldc + bc] += A[(size_t)i * lda + ac];
}

// Final step: reduce constraints against the input box, refine ub/lb in place
__global__ void k_reduce_input(const float* __restrict__ Fu, const float* __restrict__ Fl,
                               int ld, const float* __restrict__ ub0, const float* __restrict__ lb0,
                               float* __restrict__ ub, float* __restrict__ lb)
{
    __shared__ float r1[256], r2[256];
    int row = blockIdx.x, tid = threadIdx.x;
    const float* fu = Fu + (size_t)row * ld;
    const float* fl = Fl + (size_t)row * ld;
    float au = 0.0f, al = 0.0f;
    for (int j = tid; j < DIN; j += 256) {
        float u = fu[j];
        au += (u > 0.0f) ? u * ub0[j] : u * lb0[j];
        float l = fl[j];
        al += (l > 0.0f) ? l * lb0[j] : l * ub0[j];
    }
    r1[tid] = au; r2[tid] = al; __syncthreads();
    for (int s = 128; s > 0; s >>= 1) {
        if (tid < s) { r1[tid] += r1[tid + s]; r2[tid] += r2[tid + s]; }
        __syncthreads();
    }
    if (tid == 0) {
        ub[row] = fminf(ub[row], r1[0] + fu[BC1]);
        lb[row] = fmaxf(lb[row], r2[0] + fl[BC1]);
    }
}

__global__ void k_final_min(const float* __restrict__ lb, int n, float* __restrict__ out)
{
    float m = lb[0];
    for (int i = 1; i < n; ++i) m = fminf(m, lb[i]);
    out[0] = m;
}

// ---------------------------------------------------------------- driver ----
static inline void launch_gemm(const float* A, int lda, const float* P, int ldp,
                               float* C, int M, hipStream_t s)
{
    dim3 grid(ldp / GEMM_TN, M / GEMM_TM);
    wmma_gemm_kernel<<<grid, 128, 0, s>>>(A, lda, P, ldp, C, ldp, H);
}

extern "C" void kernel_launch(void* const* d_in, const int* in_sizes, int n_in,
                              void* d_out, int out_size, void* d_ws, size_t ws_size,
                              hipStream_t stream)
{
    const float* inp = (const float*)d_in[0];
    const float* eps = (const float*)d_in[1];
    const float* w1  = (const float*)d_in[2];
    const float* b1  = (const float*)d_in[3];
    const float* w2  = (const float*)d_in[4];
    const float* b2  = (const float*)d_in[5];
    const float* w3  = (const float*)d_in[6];
    const float* b3  = (const float*)d_in[7];
    const float* w4  = (const float*)d_in[8];
    const float* b4  = (const float*)d_in[9];
    float* out = (float*)d_out;

    float* wsp = (float*)d_ws;
    size_t off = 0;
    auto alloc = [&](size_t n) { float* p = wsp + off; off += (n + 63) & ~(size_t)63; return p; };

    float* ub0 = alloc(DIN); float* lb0 = alloc(DIN);
    float* u1 = alloc(H); float* l1 = alloc(H);
    float* su1 = alloc(H); float* iu1 = alloc(H); float* sl1 = alloc(H); float* il1 = alloc(H);
    float* us1 = alloc(H); float* ls1 = alloc(H);
    float* u2 = alloc(H); float* l2 = alloc(H);
    float* su2 = alloc(H); float* iu2 = alloc(H); float* sl2 = alloc(H); float* il2 = alloc(H);
    float* us2 = alloc(H); float* ls2 = alloc(H);
    float* u3 = alloc(H); float* l3 = alloc(H);
    float* su3 = alloc(H); float* iu3 = alloc(H); float* sl3 = alloc(H); float* il3 = alloc(H);
    float* us3 = alloc(H); float* ls3 = alloc(H);
    float* u4 = alloc(64); float* l4 = alloc(64);
    float* c1p = alloc((size_t)H * LD1);
    float* c2p = alloc((size_t)H * LD2);
    float* c3p = alloc((size_t)H * LD2);
    float* Xu = alloc((size_t)H * LD2); float* Xl = alloc((size_t)H * LD2);
    float* Yu = alloc((size_t)H * LD2); float* Yl = alloc((size_t)H * LD2);

    const size_t eTot1 = (size_t)H * LD1;
    const size_t eTot2 = (size_t)H * LD2;

    k_input_bounds<<<(DIN + 255) / 256, 256, 0, stream>>>(inp, eps, ub0, lb0);
    k_build_c<<<(unsigned)((eTot1 + 255) / 256), 256, 0, stream>>>(w1, b1, c1p, H, DIN, LD1);
    k_build_c<<<(unsigned)((eTot2 + 255) / 256), 256, 0, stream>>>(w2, b2, c2p, H, H, LD2);
    k_build_c<<<(unsigned)((eTot2 + 255) / 256), 256, 0, stream>>>(w3, b3, c3p, H, H, LD2);

    // ---- layer 1 affine (chain == input, interval bound is exact) ----
    k_interval_affine<<<H, 256, 0, stream>>>(w1, b1, ub0, lb0, DIN, u1, l1);
    // ---- spu 1 ----
    k_spu_analyze<<<(H + 255) / 256, 256, 0, stream>>>(u1, l1, su1, iu1, sl1, il1, us1, ls1, H);
    k_bs_spu1<<<H, 256, 0, stream>>>(w1, b1, su1, iu1, sl1, il1, ub0, lb0, us1, ls1);

    // ---- layer 2 affine: c2 -> spu1 -> c1 -> input ----
    k_interval_affine<<<H, 256, 0, stream>>>(w2, b2, us1, ls1, H, u2, l2);
    k_diag_from_affine<<<H, 256, 0, stream>>>(w2, b2, su1, iu1, sl1, il1, Xu, Xl, LD2, H);
    launch_gemm(Xu, LD2, c1p, LD1, Yu, H, stream);
    launch_gemm(Xl, LD2, c1p, LD1, Yl, H, stream);
    k_add_bias<<<(H + 255) / 256, 256, 0, stream>>>(Yu, LD1, BC1, Xu, LD2, BC2, H);
    k_add_bias<<<(H + 255) / 256, 256, 0, stream>>>(Yl, LD1, BC1, Xl, LD2, BC2, H);
    k_reduce_input<<<H, 256, 0, stream>>>(Yu, Yl, LD1, ub0, lb0, u2, l2);

    // ---- spu 2: diag -> c2 -> spu1 -> c1 -> input ----
    k_spu_analyze<<<(H + 255) / 256, 256, 0, stream>>>(u2, l2, su2, iu2, sl2, il2, us2, ls2, H);
    k_rowscale<<<(unsigned)((eTot2 + 255) / 256), 256, 0, stream>>>(c2p, su2, iu2, sl2, il2, Xu, Xl, LD2, BC2, eTot2);
    k_colpass<<<H, 256, 0, stream>>>(Xu, Xl, su1, iu1, sl1, il1, Yu, Yl, LD2);
    launch_gemm(Yu, LD2, c1p, LD1, Xu, H, stream);
    launch_gemm(Yl, LD2, c1p, LD1, Xl, H, stream);
    k_add_bias<<<(H + 255) / 256, 256, 0, stream>>>(Xu, LD1, BC1, Yu, LD2, BC2, H);
    k_add_bias<<<(H + 255) / 256, 256, 0, stream>>>(Xl, LD1, BC1, Yl, LD2, BC2, H);
    k_reduce_input<<<H, 256, 0, stream>>>(Xu, Xl, LD1, ub0, lb0, us2, ls2);

    // ---- layer 3 affine: c3 -> spu2 -> c2 -> spu1 -> c1 -> input ----
    k_interval_affine<<<H, 256, 0, stream>>>(w3, b3, us2, ls2, H, u3, l3);
    k_diag_from_affine<<<H, 256, 0, stream>>>(w3, b3, su2, iu2, sl2, il2, Xu, Xl, LD2, H);
    launch_gemm(Xu, LD2, c2p, LD2, Yu, H, stream);
    launch_gemm(Xl, LD2, c2p, LD2, Yl, H, stream);
    k_add_bias<<<(H + 255) / 256, 256, 0, stream>>>(Yu, LD2, BC2, Xu, LD2, BC2, H);
    k_add_bias<<<(H + 255) / 256, 256, 0, stream>>>(Yl, LD2, BC2, Xl, LD2, BC2, H);
    k_colpass<<<H, 256, 0, stream>>>(Yu, Yl, su1, iu1, sl1, il1, Xu, Xl, LD2);
    launch_gemm(Xu, LD2, c1p, LD1, Yu, H, stream);
    launch_gemm(Xl, LD2, c1p, LD1, Yl, H, stream);
    k_add_bias<<<(H + 255) / 256, 256, 0, stream>>>(Yu, LD1, BC1, Xu, LD2, BC2, H);
    k_add_bias<<<(H + 255) / 256, 256, 0, stream>>>(Yl, LD1, BC1, Xl, LD2, BC2, H);
    k_reduce_input<<<H, 256, 0, stream>>>(Yu, Yl, LD1, ub0, lb0, u3, l3);

    // ---- spu 3: diag -> c3 -> spu2 -> c2 -> spu1 -> c1 -> input ----
    k_spu_analyze<<<(H + 255) / 256, 256, 0, stream>>>(u3, l3, su3, iu3, sl3, il3, us3, ls3, H);
    k_rowscale<<<(unsigned)((eTot2 + 255) / 256), 256, 0, stream>>>(c3p, su3, iu3, sl3, il3, Xu, Xl, LD2, BC2, eTot2);
    k_colpass<<<H, 256, 0, stream>>>(Xu, Xl, su2, iu2, sl2, il2, Yu, Yl, LD2);
    launch_gemm(Yu, LD2, c2p, LD2, Xu, H, stream);
    launch_gemm(Yl, LD2, c2p, LD2, Xl, H, stream);
    k_add_bias<<<(H + 255) / 256, 256, 0, stream>>>(Xu, LD2, BC2, Yu, LD2, BC2, H);
    k_add_bias<<<(H + 255) / 256, 256, 0, stream>>>(Xl, LD2, BC2, Yl, LD2, BC2, H);
    k_colpass<<<H, 256, 0, stream>>>(Xu, Xl, su1, iu1, sl1, il1, Yu, Yl, LD2);
    launch_gemm(Yu, LD2, c1p, LD1, Xu, H, stream);
    launch_gemm(Yl, LD2, c1p, LD1, Xl, H, stream);
    k_add_bias<<<(H + 255) / 256, 256, 0, stream>>>(Xu, LD1, BC1, Yu, LD2, BC2, H);
    k_add_bias<<<(H + 255) / 256, 256, 0, stream>>>(Xl, LD1, BC1, Yl, LD2, BC2, H);
    k_reduce_input<<<H, 256, 0, stream>>>(Xu, Xl, LD1, ub0, lb0, us3, ls3);

    // ---- layer 4 affine (10 rows, padded to 64): c4 -> spu3 -> c3 -> spu2 -> c2 -> spu1 -> c1 -> input ----
    k_interval_affine<<<DOUT, 256, 0, stream>>>(w4, b4, us3, ls3, H, u4, l4);
    k_diag_from_affine<<<64, 256, 0, stream>>>(w4, b4, su3, iu3, sl3, il3, Xu, Xl, LD2, DOUT);
    launch_gemm(Xu, LD2, c3p, LD2, Yu, 64, stream);
    launch_gemm(Xl, LD2, c3p, LD2, Yl, 64, stream);
    k_add_bias<<<1, 64, 0, stream>>>(Yu, LD2, BC2, Xu, LD2, BC2, 64);
    k_add_bias<<<1, 64, 0, stream>>>(Yl, LD2, BC2, Xl, LD2, BC2, 64);
    k_colpass<<<64, 256, 0, stream>>>(Yu, Yl, su2, iu2, sl2, il2, Xu, Xl, LD2);
    launch_gemm(Xu, LD2, c2p, LD2, Yu, 64, stream);
    launch_gemm(Xl, LD2, c2p, LD2, Yl, 64, stream);
    k_add_bias<<<1, 64, 0, stream>>>(Yu, LD2, BC2, Xu, LD2, BC2, 64);
    k_add_bias<<<1, 64, 0, stream>>>(Yl, LD2, BC2, Xl, LD2, BC2, 64);
    k_colpass<<<64, 256, 0, stream>>>(Yu, Yl, su1, iu1, sl1, il1, Xu, Xl, LD2);
    launch_gemm(Xu, LD2, c1p, LD1, Yu, 64, stream);
    launch_gemm(Xl, LD2, c1p, LD1, Yl, 64, stream);
    k_add_bias<<<1, 64, 0, stream>>>(Yu, LD1, BC1, Xu, LD2, BC2, 64);
    k_add_bias<<<1, 64, 0, stream>>>(Yl, LD1, BC1, Xl, LD2, BC2, 64);
    k_reduce_input<<<DOUT, 256, 0, stream>>>(Yu, Yl, LD1, ub0, lb0, u4, l4);

    k_final_min<<<1, 1, 0, stream>>>(l4, DOUT, out);
}